// GNN_24876450578861
// MI455X (gfx1250) — compile-verified
//
#include <hip/hip_runtime.h>

#define NUM_USERS 100000
#define NUM_ITEMS 50000
#define NN        (NUM_USERS + NUM_ITEMS)   // 150000
#define DIM       64
#define NE        2000000
#define NELEMS    (NN * DIM)                // 9,600,000

// d_out flat layout (floats):
//  [0,            6,400,000)  emb_users_final
//  [6,400,000,   12,800,000)  emb_users (copy of input)
//  [12,800,000,  16,000,000)  emb_items_final
//  [16,000,000,  19,200,000)  emb_items (copy of input)
#define OUT_U_COPY  (NUM_USERS * DIM)                       // 6,400,000
#define OUT_I_FINAL (2 * NUM_USERS * DIM)                   // 12,800,000
#define OUT_I_COPY  (2 * NUM_USERS * DIM + NUM_ITEMS * DIM) // 16,000,000

// acc(node, d) lives in the *final* output regions (split layout):
//   node < NUM_USERS : out[node*64 + d]
//   else             : out[OUT_U_COPY + node*64 + d]   (== OUT_I_FINAL + (node-NU)*64 + d)

typedef __attribute__((ext_vector_type(2))) float v2f;
typedef __attribute__((ext_vector_type(8))) float v8f;

// init: cur = concat(users, items); acc = same (in out split layout);
// also emit the verbatim input copies into out regions 1 and 3.
__global__ void gnn_init(const float* __restrict__ users,
                         const float* __restrict__ items,
                         float* __restrict__ cur,
                         float* __restrict__ out) {
    int i = blockIdx.x * blockDim.x + threadIdx.x;
    if (i >= NELEMS) return;
    int node = i >> 6;
    float e;
    int accoff, copyoff;
    if (node < NUM_USERS) {
        e = users[i];
        accoff  = i;
        copyoff = OUT_U_COPY + i;
    } else {
        e = items[i - NUM_USERS * DIM];
        accoff  = OUT_U_COPY + i;                     // lands in [12.8M, 16M)
        copyoff = OUT_I_COPY + (i - NUM_USERS * DIM); // lands in [16M, 19.2M)
    }
    cur[i]       = e;
    out[accoff]  = e;
    out[copyoff] = e;
}

__global__ void gnn_zero(float* __restrict__ p) {
    int i = blockIdx.x * blockDim.x + threadIdx.x;
    if (i < NELEMS) p[i] = 0.0f;
}

// 16 threads per edge, 4 dims each (float4 gather + 4 native f32 atomics).
__global__ void gnn_scatter(const int* __restrict__ eidx,
                            const float* __restrict__ ew,
                            const float* __restrict__ cur,
                            float* __restrict__ nxt) {
    long long t = (long long)blockIdx.x * blockDim.x + threadIdx.x;
    int e = (int)(t >> 4);
    if (e >= NE) return;
    int d = ((int)t & 15) << 2;
    int src = eidx[e];
    int dst = eidx[NE + e];
    float w = ew[e];
    const float4 v = *reinterpret_cast<const float4*>(cur + (size_t)src * DIM + d);
    float* p = nxt + (size_t)dst * DIM + d;
    unsafeAtomicAdd(p + 0, w * v.x);   // global_atomic_add_f32
    unsafeAtomicAdd(p + 1, w * v.y);
    unsafeAtomicAdd(p + 2, w * v.z);
    unsafeAtomicAdd(p + 3, w * v.w);
}

// relu in place; acc += relu(nxt)  (acc lives in out, split layout)
__global__ void gnn_relu_acc(float* __restrict__ nxt, float* __restrict__ out) {
    int i = blockIdx.x * blockDim.x + threadIdx.x;
    if (i >= NELEMS) return;
    float v = nxt[i];
    v = v > 0.0f ? v : 0.0f;
    nxt[i] = v;
    int node = i >> 6;
    int off = (node < NUM_USERS) ? i : (OUT_U_COPY + i);
    out[off] += v;
}

// Final dense layer, fp32 WMMA 16x16x4, in place on the acc/out regions.
// One wave per 16-row tile; 4 n-tiles cover all 64 output dims; K=64 in 16 steps.
// ISA 7.12.2 fp32 A layout: lanes 0-15 -> M rows, v2f.x=K+0,.y=K+1 (half 0),
//                           lanes 16-31 -> same rows, K+2/K+3 (half 1).
// B[k][n] = W[n][k]  ->  contiguous v2f load from W + n*64 + k.
__global__ void gnn_gemm(const float* __restrict__ W,
                         const float* __restrict__ bias,
                         float* __restrict__ out) {
    int tile = blockIdx.x;               // 0 .. 9374 (exact grid; EXEC all ones)
    int lane = threadIdx.x;              // wave32
    int row0 = tile * 16;
    float* base = out + ((row0 < NUM_USERS) ? (size_t)row0 * DIM
                                            : (size_t)OUT_U_COPY + (size_t)row0 * DIM);
    int half = lane >> 4;                // 0: K+0/K+1 lanes, 1: K+2/K+3 lanes
    int m    = lane & 15;

    v8f c0 = {}, c1 = {}, c2 = {}, c3 = {};
    const float scale = 1.0f / 3.0f;     // acc / (NUM_LAYERS + 1)

    for (int k = 0; k < 16; ++k) {
        int kk = (k << 2) + (half << 1);
        v2f a = *reinterpret_cast<const v2f*>(base + m * DIM + kk);
        a = a * scale;
        v2f b0 = *reinterpret_cast<const v2f*>(W + ( 0 + m) * DIM + kk);
        v2f b1 = *reinterpret_cast<const v2f*>(W + (16 + m) * DIM + kk);
        v2f b2 = *reinterpret_cast<const v2f*>(W + (32 + m) * DIM + kk);
        v2f b3 = *reinterpret_cast<const v2f*>(W + (48 + m) * DIM + kk);
        c0 = __builtin_amdgcn_wmma_f32_16x16x4_f32(false, a, false, b0, (short)0, c0, false, false);
        c1 = __builtin_amdgcn_wmma_f32_16x16x4_f32(false, a, false, b1, (short)0, c1, false, false);
        c2 = __builtin_amdgcn_wmma_f32_16x16x4_f32(false, a, false, b2, (short)0, c2, false, false);
        c3 = __builtin_amdgcn_wmma_f32_16x16x4_f32(false, a, false, b3, (short)0, c3, false, false);
    }

    float bb0 = bias[ 0 + m], bb1 = bias[16 + m], bb2 = bias[32 + m], bb3 = bias[48 + m];

    // D layout: VGPR v -> row v (lanes 0-15) / row v+8 (lanes 16-31), col = lane&15.
    for (int v = 0; v < 8; ++v) {
        int r = v + (half << 3);
        float* p = base + r * DIM + m;
        p[ 0] = c0[v] + bb0;
        p[16] = c1[v] + bb1;
        p[32] = c2[v] + bb2;
        p[48] = c3[v] + bb3;
    }
}

extern "C" void kernel_launch(void* const* d_in, const int* in_sizes, int n_in,
                              void* d_out, int out_size, void* d_ws, size_t ws_size,
                              hipStream_t stream) {
    const int*   eidx  = (const int*)d_in[0];    // edge_index (2, E)
    const float* ew    = (const float*)d_in[1];  // edge_weight (E,)
    const float* users = (const float*)d_in[2];  // (NUM_USERS, 64)
    const float* items = (const float*)d_in[3];  // (NUM_ITEMS, 64)
    const float* W     = (const float*)d_in[4];  // (64, 64)
    const float* bias  = (const float*)d_in[5];  // (64,)
    float* out = (float*)d_out;

    float* cur = (float*)d_ws;                   // 38.4 MB
    float* nxt = cur + (size_t)NELEMS;           // 38.4 MB

    const int tpb = 256;
    const int gElem = (NELEMS + tpb - 1) / tpb;

    gnn_init<<<gElem, tpb, 0, stream>>>(users, items, cur, out);

    for (int layer = 0; layer < 2; ++layer) {
        gnn_zero<<<gElem, tpb, 0, stream>>>(nxt);
        gnn_scatter<<<(NE * 16) / tpb, tpb, 0, stream>>>(eidx, ew, cur, nxt);
        gnn_relu_acc<<<gElem, tpb, 0, stream>>>(nxt, out);
        float* t = cur; cur = nxt; nxt = t;
    }

    // 9375 one-wave workgroups (150000/16), EXEC all ones for WMMA.
    gnn_gemm<<<NN / 16, 32, 0, stream>>>(W, bias, out);
}